// SplineBlock_6828998001341
// MI455X (gfx1250) — compile-verified
//
#include <hip/hip_runtime.h>
#include <math.h>

// ---------------------------------------------------------------------------
// SplineCNN block for MI455X (gfx1250, wave32).
//   T[n,k,:] = x[n] @ W[k]  : V_WMMA_F32_16X16X4_F32, A frags register-resident
//                             (loaded once per node tile), B frags as single
//                             ds_load_b64 from a fragment-ordered LDS slab
//                             (conflict-free: halves tile banks 0/1 vs 2/3 mod 4)
//   edge pass               : LDS-staged basis/widx, 8 coalesced 128B taps of
//                             T[src], one f32 atomic per (dst, channel)
//   finalize                : acc/deg + x@root (WMMA) + bias (+ELU)
//   BN + ELU + concat(pos)  : two-pass reduction, fused apply
// Workspace requirement ~235 MB (T buffer reused across layer-2 out-chunks).
// ---------------------------------------------------------------------------

static constexpr int NN   = 50000;   // nodes
static constexpr int NE   = 400000;  // edges
static constexpr int KW   = 27;      // spline kernel count
static constexpr int IN1  = 32;
static constexpr int MID  = 32;
static constexpr int OUT2 = 64;
static constexpr int C3P  = 68;      // 64 + 3 (pos) padded to 68
static constexpr int OUTF = 32;
static constexpr int NTILE = NN / 16;  // 3125 (exact)

typedef float v2f __attribute__((ext_vector_type(2)));
typedef float v8f __attribute__((ext_vector_type(8)));

__device__ __forceinline__ float elu1(float v) { return v > 0.f ? v : expm1f(v); }

// Fully-unrolled 16x16 f32 tile from global A/B (used by finalize's root GEMM).
template <int KD>
__device__ __forceinline__ v8f wmma_tile_g(const float* __restrict__ A, int lda,
                                           const float* __restrict__ B, int ldb,
                                           int lane) {
  v8f c = {};
  const int half = lane >> 4;
  const int idx  = lane & 15;
#pragma unroll
  for (int kk = 0; kk < KD; kk += 4) {
    const int k0 = kk + 2 * half;
    v2f a = *reinterpret_cast<const v2f*>(A + idx * lda + k0);  // 8B aligned
    v2f b;
    b.x = B[k0 * ldb + idx];
    b.y = B[(k0 + 1) * ldb + idx];
    c = __builtin_amdgcn_wmma_f32_16x16x4_f32(false, a, false, b, (short)0, c,
                                              false, false);
  }
  return c;
}

// ---------------------------------------------------------------------------
// 1) Per-edge degree + degree-1 closed B-spline basis (8 taps into K=27).
// ---------------------------------------------------------------------------
__global__ void k_basis(const float* __restrict__ attr, const int* __restrict__ dst,
                        float* __restrict__ basis, int* __restrict__ widx,
                        float* __restrict__ deg) {
  int e = blockIdx.x * blockDim.x + threadIdx.x;
  if (e >= NE) return;
  float fr[3]; int lo[3];
#pragma unroll
  for (int d = 0; d < 3; ++d) {
    float s  = attr[e * 3 + d] * 3.0f;
    float fl = floorf(s);
    lo[d] = (int)fl;
    fr[d] = s - fl;
  }
#pragma unroll
  for (int b = 0; b < 8; ++b) {
    float p = 1.f; int w = 0, stride = 1;
#pragma unroll
    for (int d = 0; d < 3; ++d) {
      int bit = (b >> d) & 1;
      p *= bit ? fr[d] : (1.f - fr[d]);
      int id = lo[d] + bit;
      if (id >= 3) id -= 3;
      w += id * stride;
      stride *= 3;
    }
    basis[e * 8 + b] = p;
    widx[e * 8 + b]  = w;
  }
  atomicAdd(&deg[dst[e]], 1.0f);
}

// ---------------------------------------------------------------------------
// 2) Zero-pad W3 [27,67,32] -> [27,68,32] and root3 [67,32] -> [68,32].
// ---------------------------------------------------------------------------
__global__ void k_pad3(const float* __restrict__ W3, const float* __restrict__ root3,
                       float* __restrict__ W3p, float* __restrict__ root3p) {
  int gid = blockIdx.x * blockDim.x + threadIdx.x;
  const int totW = KW * C3P * OUTF;  // 58752
  if (gid < totW) {
    int o = gid % OUTF;
    int i = (gid / OUTF) % C3P;
    int k = gid / (OUTF * C3P);
    W3p[gid] = (i < 67) ? W3[(k * 67 + i) * OUTF + o] : 0.f;
  } else if (gid < totW + C3P * OUTF) {
    int g = gid - totW;
    int o = g % OUTF;
    int i = g / OUTF;
    root3p[g] = (i < 67) ? root3[i * OUTF + o] : 0.f;
  }
}

// ---------------------------------------------------------------------------
// 3) Dense lift: T[n, k, oc(0..31)] = X[n,:] @ W[k][:, outBase+oc].
//    Block = 8 waves, one node tile per wave. A fragments live in registers
//    for the whole kernel. W is staged 4 k's at a time into LDS in FRAGMENT
//    ORDER: element (row i = 4s+2h+j, col oc) lives at
//        Wlds[((kL*KSTEPS + s)*32 + oc)*4 + h*2 + j]
//    so each lane's B fragment is ONE aligned ds_load_b64, and the wave's 32
//    b64 accesses tile all 64 LDS banks exactly once (no conflicts, no movs).
// ---------------------------------------------------------------------------
template <int INDIM>
__global__ void __launch_bounds__(256)
k_gemm_T(const float* __restrict__ X, const float* __restrict__ W,
         float* __restrict__ T, int outDim, int outBase) {
  constexpr int KSTEPS = INDIM / 4;
  __shared__ float Wlds[4 * KSTEPS * 32 * 4];  // = 4*INDIM*32 floats

  const int wave = threadIdx.x >> 5;
  const int lane = threadIdx.x & 31;
  const int nodeTile = blockIdx.x * 8 + wave;   // wave-uniform
  const int half = lane >> 4;
  const int idx  = lane & 15;

  v2f areg[KSTEPS];
  if (nodeTile < NTILE) {
    const float* A = X + ((size_t)nodeTile * 16 + idx) * INDIM + 2 * half;
#pragma unroll
    for (int s = 0; s < KSTEPS; ++s)
      areg[s] = *reinterpret_cast<const v2f*>(A + 4 * s);  // 8B aligned
  }

  for (int kBase = 0; kBase < KW; kBase += 4) {
    const int kc  = (KW - kBase) < 4 ? (KW - kBase) : 4;
    const int tot = kc * INDIM * 32;
    for (int t = threadIdx.x; t < tot; t += 256) {
      int kL  = t / (INDIM * 32);
      int rem = t - kL * INDIM * 32;
      int i   = rem >> 5;   // W row (reduction index)
      int oc  = rem & 31;   // W col within 32-chunk
      // fragment-order scatter: s = i>>2, h = (i>>1)&1, j = i&1
      Wlds[((kL * KSTEPS + (i >> 2)) * 32 + oc) * 4 + ((i >> 1) & 1) * 2 + (i & 1)] =
          W[((size_t)(kBase + kL) * INDIM + i) * outDim + outBase + oc];
    }
    __syncthreads();

    if (nodeTile < NTILE) {
      for (int kL = 0; kL < kc; ++kL) {
#pragma unroll
        for (int outTile = 0; outTile < 2; ++outTile) {
          const int col = outTile * 16 + idx;
          const float* Bl = &Wlds[(kL * KSTEPS * 32 + col) * 4 + half * 2];
          v8f c = {};
#pragma unroll
          for (int s = 0; s < KSTEPS; ++s) {
            v2f b = *reinterpret_cast<const v2f*>(Bl + s * 128);  // ds_load_b64
            c = __builtin_amdgcn_wmma_f32_16x16x4_f32(false, areg[s], false, b,
                                                      (short)0, c, false, false);
          }
          const int k = kBase + kL;
          float* Tp = T + (size_t)(nodeTile * 16 + 8 * half) * (KW * 32) + k * 32 + col;
#pragma unroll
          for (int r = 0; r < 8; ++r)
            Tp[(size_t)r * (KW * 32)] = c[r];
        }
      }
    }
    __syncthreads();
  }
}

// ---------------------------------------------------------------------------
// 4) Edge aggregation: block = 8 edges x 32 channels. basis/widx staged via
//    LDS (broadcast reads), 8 coalesced 128B taps of T[src], one atomic.
// ---------------------------------------------------------------------------
__global__ void __launch_bounds__(256)
k_edge_agg(const float* __restrict__ T, const float* __restrict__ basis,
           const int* __restrict__ widx, const int* __restrict__ src,
           const int* __restrict__ dst, float* __restrict__ acc,
           int accStride, int accBase) {
  __shared__ float bsh[64];
  __shared__ int   ksh[64];
  const int eBase = blockIdx.x * 8;
  const int tid   = threadIdx.x;
  if (tid < 64) {
    bsh[tid] = basis[(size_t)eBase * 8 + tid];
    ksh[tid] = widx [(size_t)eBase * 8 + tid];
  }
  __syncthreads();

  const int eL = tid >> 5;
  const int oc = tid & 31;
  const int e  = eBase + eL;
  const int s  = src[e];
  const int d  = dst[e];
  const float* Ts = T + (size_t)s * (KW * 32) + oc;
  float sum = 0.f;
#pragma unroll
  for (int b = 0; b < 8; ++b)
    sum += bsh[eL * 8 + b] * Ts[ksh[eL * 8 + b] * 32];
  atomicAdd(&acc[(size_t)d * accStride + accBase + oc], sum);
}

// ---------------------------------------------------------------------------
// 5) Finalize: out = acc/deg + X@root + bias (+ ELU).  WMMA for X@root.
// ---------------------------------------------------------------------------
template <int INDIM>
__global__ void k_finalize(const float* __restrict__ X, const float* __restrict__ root,
                           const float* __restrict__ bias, const float* __restrict__ acc,
                           const float* __restrict__ deg, float* __restrict__ out,
                           int outDim, int applyElu, int nTiles) {
  int waveId = (int)((blockIdx.x * blockDim.x + threadIdx.x) >> 5);
  int lane   = threadIdx.x & 31;
  if (waveId >= nTiles) return;  // wave-uniform
  int outTiles = outDim >> 4;
  int outTile  = waveId % outTiles;
  int nodeTile = waveId / outTiles;

  const float* A = X + (size_t)nodeTile * 16 * INDIM;
  const float* B = root + outTile * 16;
  v8f c = wmma_tile_g<INDIM>(A, INDIM, B, outDim, lane);

  const int half = lane >> 4;
  const int col  = (lane & 15) + outTile * 16;
  const float bc = bias[col];
#pragma unroll
  for (int r = 0; r < 8; ++r) {
    int row = nodeTile * 16 + r + 8 * half;
    float dg = fmaxf(deg[row], 1.f);
    float v  = c[r] + acc[(size_t)row * outDim + col] / dg + bc;
    if (applyElu) v = elu1(v);
    out[(size_t)row * outDim + col] = v;
  }
}

// ---------------------------------------------------------------------------
// 6) BatchNorm statistics (64 channels over N), block = 64 nodes x 64 ch.
// ---------------------------------------------------------------------------
__global__ void k_bn_stats(const float* __restrict__ z, float* __restrict__ sums,
                           float* __restrict__ sumsq) {
  __shared__ float ls[256], lq[256];
  int c   = threadIdx.x & 63;
  int sub = threadIdx.x >> 6;  // 0..3
  int nodeBase = blockIdx.x * 64;
  float s = 0.f, q = 0.f;
  for (int i = sub; i < 64; i += 4) {
    int n = nodeBase + i;
    if (n < NN) {
      float v = z[(size_t)n * OUT2 + c];
      s += v;
      q += v * v;
    }
  }
  ls[threadIdx.x] = s;
  lq[threadIdx.x] = q;
  __syncthreads();
  if (threadIdx.x < 64) {
    float ts = ls[c] + ls[c + 64] + ls[c + 128] + ls[c + 192];
    float tq = lq[c] + lq[c + 64] + lq[c + 128] + lq[c + 192];
    atomicAdd(&sums[c], ts);
    atomicAdd(&sumsq[c], tq);
  }
}

// ---------------------------------------------------------------------------
// 7) Apply BN + ELU, concat pos, zero-pad col 67 -> hcat [N, 68].
// ---------------------------------------------------------------------------
__global__ void k_bn_apply_cat(const float* __restrict__ z, const float* __restrict__ pos,
                               const float* __restrict__ sums, const float* __restrict__ sumsq,
                               const float* __restrict__ gamma, const float* __restrict__ beta,
                               float* __restrict__ hcat) {
  long gid = (long)blockIdx.x * blockDim.x + threadIdx.x;
  if (gid >= (long)NN * C3P) return;
  int n = (int)(gid / C3P);
  int c = (int)(gid % C3P);
  float v;
  if (c < OUT2) {
    const float invN = 1.0f / (float)NN;
    float mu  = sums[c] * invN;
    float var = sumsq[c] * invN - mu * mu;
    float sc  = gamma[c] * rsqrtf(var + 1e-5f);
    v = elu1((z[(size_t)n * OUT2 + c] - mu) * sc + beta[c]);
  } else if (c < 67) {
    v = pos[(size_t)n * 3 + (c - 64)];
  } else {
    v = 0.f;
  }
  hcat[gid] = v;
}

// ---------------------------------------------------------------------------
extern "C" void kernel_launch(void* const* d_in, const int* in_sizes, int n_in,
                              void* d_out, int out_size, void* d_ws, size_t ws_size,
                              hipStream_t stream) {
  const float* x     = (const float*)d_in[0];
  const int*   ei    = (const int*)d_in[1];   // [2, E]
  const float* attr  = (const float*)d_in[2];
  const float* pos   = (const float*)d_in[3];
  const float* W1    = (const float*)d_in[4];
  const float* root1 = (const float*)d_in[5];
  const float* b1    = (const float*)d_in[6];
  const float* W2    = (const float*)d_in[7];
  const float* root2 = (const float*)d_in[8];
  const float* b2    = (const float*)d_in[9];
  const float* gamma = (const float*)d_in[10];
  const float* beta  = (const float*)d_in[11];
  const float* W3    = (const float*)d_in[12];
  const float* root3 = (const float*)d_in[13];
  const float* b3    = (const float*)d_in[14];
  float* out = (float*)d_out;

  const int* srcI = ei;
  const int* dstI = ei + NE;

  // ---- workspace carve-up (256B aligned) ----
  char* p = (char*)d_ws;
  auto take = [&](size_t bytes) {
    char* r = p;
    p += (bytes + 255) & ~(size_t)255;
    return r;
  };
  float* basis  = (float*)take((size_t)NE * 8 * 4);
  int*   widx   = (int*)  take((size_t)NE * 8 * 4);
  float* deg    = (float*)take((size_t)NN * 4);
  float* T      = (float*)take((size_t)NN * KW * 32 * 4);  // 172.8 MB, reused
  float* acc    = (float*)take((size_t)NN * OUT2 * 4);
  float* h1     = (float*)take((size_t)NN * MID * 4);
  float* z2     = (float*)take((size_t)NN * OUT2 * 4);
  float* hcat   = (float*)take((size_t)NN * C3P * 4);
  float* W3p    = (float*)take((size_t)KW * C3P * OUTF * 4);
  float* root3p = (float*)take((size_t)C3P * OUTF * 4);
  float* sums   = (float*)take(2 * 64 * 4);  // sums[64] ++ sumsq[64]
  float* sumsq  = sums + 64;
  (void)ws_size; (void)n_in; (void)in_sizes; (void)out_size;

  const int BLK = 256;
  const int blkGemm = (NTILE + 7) / 8;  // 391 blocks, 8 node tiles each
  const int blkAgg  = NE / 8;           // 50000 blocks, 8 edges each

  // basis / degree / padded weights
  hipMemsetAsync(deg, 0, (size_t)NN * 4, stream);
  k_basis<<<(NE + BLK - 1) / BLK, BLK, 0, stream>>>(attr, dstI, basis, widx, deg);
  {
    int tot = KW * C3P * OUTF + C3P * OUTF;
    k_pad3<<<(tot + BLK - 1) / BLK, BLK, 0, stream>>>(W3, root3, W3p, root3p);
  }

  // ---- layer 1: 32 -> 32, ELU ----
  k_gemm_T<IN1><<<blkGemm, BLK, 0, stream>>>(x, W1, T, MID, 0);
  hipMemsetAsync(acc, 0, (size_t)NN * OUT2 * 4, stream);
  k_edge_agg<<<blkAgg, BLK, 0, stream>>>(T, basis, widx, srcI, dstI, acc, MID, 0);
  {
    int nt = NTILE * (MID / 16);
    k_finalize<IN1><<<(nt * 32 + BLK - 1) / BLK, BLK, 0, stream>>>(
        x, root1, b1, acc, deg, h1, MID, /*elu=*/1, nt);
  }

  // ---- layer 2: 32 -> 64 (two 32-col chunks through the same T buffer) ----
  hipMemsetAsync(acc, 0, (size_t)NN * OUT2 * 4, stream);
  k_gemm_T<MID><<<blkGemm, BLK, 0, stream>>>(h1, W2, T, OUT2, 0);
  k_edge_agg<<<blkAgg, BLK, 0, stream>>>(T, basis, widx, srcI, dstI, acc, OUT2, 0);
  k_gemm_T<MID><<<blkGemm, BLK, 0, stream>>>(h1, W2, T, OUT2, 32);
  k_edge_agg<<<blkAgg, BLK, 0, stream>>>(T, basis, widx, srcI, dstI, acc, OUT2, 32);
  {
    int nt = NTILE * (OUT2 / 16);
    k_finalize<MID><<<(nt * 32 + BLK - 1) / BLK, BLK, 0, stream>>>(
        h1, root2, b2, acc, deg, z2, OUT2, /*elu=*/0, nt);
  }

  // ---- batchnorm + ELU + concat(pos) ----
  hipMemsetAsync(sums, 0, 2 * 64 * 4, stream);
  k_bn_stats<<<(NN + 63) / 64, BLK, 0, stream>>>(z2, sums, sumsq);
  k_bn_apply_cat<<<(int)(((long)NN * C3P + BLK - 1) / BLK), BLK, 0, stream>>>(
      z2, pos, sums, sumsq, gamma, beta, hcat);

  // ---- layer 3: 68 (padded) -> 32, output to d_out ----
  k_gemm_T<C3P><<<blkGemm, BLK, 0, stream>>>(hcat, W3p, T, OUTF, 0);
  hipMemsetAsync(acc, 0, (size_t)NN * OUT2 * 4, stream);
  k_edge_agg<<<blkAgg, BLK, 0, stream>>>(T, basis, widx, srcI, dstI, acc, OUTF, 0);
  {
    int nt = NTILE * (OUTF / 16);
    k_finalize<C3P><<<(nt * 32 + BLK - 1) / BLK, BLK, 0, stream>>>(
        hcat, root3p, b3, acc, deg, out, OUTF, /*elu=*/0, nt);
  }
}